// GNNClassifier_21320217657456
// MI455X (gfx1250) — compile-verified
//
#include <hip/hip_runtime.h>
#include <hip/hip_bf16.h>

typedef __attribute__((ext_vector_type(2))) float v2f;
typedef __attribute__((ext_vector_type(8))) float v8f;

#define N_NODES    50000
#define N_EDGES    800000
#define NUM_GRAPHS 16
#define NUM_CLASSES 247

// ---------------------------------------------------------------- utilities
__global__ void k_fill(float* __restrict__ p, float v, int n) {
    int i = blockIdx.x * blockDim.x + threadIdx.x;
    if (i < n) p[i] = v;
}

// deg[col[e]] += 1  (dinv buffer pre-filled with 1.0 for the self-loop)
__global__ void k_deg(const int* __restrict__ col, float* __restrict__ deg, int E) {
    int e = blockIdx.x * blockDim.x + threadIdx.x;
    if (e < E) atomicAdd(&deg[col[e]], 1.0f);
}

__global__ void k_rsqrt(float* __restrict__ d, int n) {
    int i = blockIdx.x * blockDim.x + threadIdx.x;
    if (i < n) d[i] = rsqrtf(d[i]);   // deg >= 1 always (self loops)
}

// ------------------------------------------------- layer 1 GEMM (K=3, tiny)
__global__ void k_gemm1(const float* __restrict__ x, const float* __restrict__ W1,
                        float* __restrict__ out, int N) {
    int t = blockIdx.x * blockDim.x + threadIdx.x;
    int n = t >> 6, f = t & 63;
    if (n < N) {
        out[t] = fmaf(x[n * 3 + 0], W1[f],
                 fmaf(x[n * 3 + 1], W1[64 + f],
                      x[n * 3 + 2] * W1[128 + f]));
    }
}

// ----------------------------------------- WMMA f32 GEMM: Hout = Hin @ W
// One wave computes a 16x16 f32 tile via V_WMMA_F32_16X16X4_F32, K step 4.
// A 16x4 layout : lanes 0-15 -> M=lane, v[0]=K0 v[1]=K1 ; lanes 16-31 -> K2,K3
// B 4x16 layout : lanes 0-15 -> N=lane, v[0]=row K0 v[1]=row K1 ; hi half K2,K3
// C/D 16x16     : lanes 0-15 N=lane M=vgpr ; lanes 16-31 N=lane-16 M=vgpr+8
__global__ void k_gemm_wmma(const float* __restrict__ Hin, const float* __restrict__ W,
                            float* __restrict__ Hout, int N, int K, int OUT) {
    int wave = (blockIdx.x * blockDim.x + threadIdx.x) >> 5;
    int lane = threadIdx.x & 31;
    int ntilesN = OUT >> 4;
    int mt = wave / ntilesN;
    int nt = wave - mt * ntilesN;
    if (mt * 16 >= N) return;                 // wave-uniform: EXEC stays all-1s
    int m0 = mt << 4, n0 = nt << 4;
    int idx16  = lane & 15;                   // M row (for A) / N col (for B,D)
    int khalf  = lane >> 4;                   // selects K pair {0,1} vs {2,3}

    v8f acc = {};
    const float* arow = Hin + (size_t)(m0 + idx16) * K + 2 * khalf;
    for (int k0 = 0; k0 < K; k0 += 4) {
        v2f a, b;
        a[0] = arow[k0];
        a[1] = arow[k0 + 1];
        const float* bp = W + (size_t)(k0 + 2 * khalf) * OUT + n0 + idx16;
        b[0] = bp[0];
        b[1] = bp[OUT];
        acc = __builtin_amdgcn_wmma_f32_16x16x4_f32(
            /*neg_a=*/false, a, /*neg_b=*/false, b,
            /*c_mod=*/(short)0, acc, /*reuse_a=*/false, /*reuse_b=*/false);
    }
#pragma unroll
    for (int r = 0; r < 8; ++r)
        Hout[(size_t)(m0 + r + 8 * khalf) * OUT + n0 + idx16] = acc[r];
}

// ---------------------- self-loop contribution: agg[n,f] = h[n,f] * dinv[n]^2
__global__ void k_selfloop(const float* __restrict__ h, const float* __restrict__ dinv,
                           float* __restrict__ agg, int total, int shF) {
    int t = blockIdx.x * blockDim.x + threadIdx.x;
    if (t < total) {
        float dv = dinv[t >> shF];
        agg[t] = h[t] * dv * dv;
    }
}

// ------------- edge scatter: agg[col] += h[row] * dinv[row]*dinv[col] (f32x4)
__global__ void k_edge(const float* __restrict__ h, const int* __restrict__ row,
                       const int* __restrict__ col, const float* __restrict__ dinv,
                       float* __restrict__ agg, int E, int shC, int F) {
    int t = blockIdx.x * blockDim.x + threadIdx.x;
    int e = t >> shC;
    if (e >= E) return;
    int c = (t & ((1 << shC) - 1)) << 2;
    int r = row[e], d = col[e];
    float nrm = dinv[r] * dinv[d];
    const float4 v = *reinterpret_cast<const float4*>(h + (size_t)r * F + c);
    float* ap = agg + (size_t)d * F + c;
    atomicAdd(ap + 0, v.x * nrm);
    atomicAdd(ap + 1, v.y * nrm);
    atomicAdd(ap + 2, v.z * nrm);
    atomicAdd(ap + 3, v.w * nrm);
}

// --------------------------------------- h = relu(h + b)  (F power of two)
__global__ void k_bias_relu(float* __restrict__ h, const float* __restrict__ b,
                            int total, int Fmask) {
    int t = blockIdx.x * blockDim.x + threadIdx.x;
    if (t < total) {
        float v = h[t] + b[t & Fmask];
        h[t] = v > 0.0f ? v : 0.0f;
    }
}

// ----------------------------------------------------- global mean pool (sum)
__global__ void k_pool(const float* __restrict__ h, const int* __restrict__ batch,
                       float* __restrict__ pooled, float* __restrict__ cnt, int N) {
    int t = blockIdx.x * blockDim.x + threadIdx.x;
    if (t >= N * 256) return;
    int n = t >> 8, f = t & 255;
    int g = batch[n];
    atomicAdd(&pooled[g * 256 + f], h[t]);
    if (f == 0) atomicAdd(&cnt[g], 1.0f);
}

// ------------------------------------------- out = (pooled/cnt) @ Wfc + bfc
__global__ void k_fc(const float* __restrict__ pooled, const float* __restrict__ cnt,
                     const float* __restrict__ Wfc, const float* __restrict__ bfc,
                     float* __restrict__ out) {
    int t = blockIdx.x * blockDim.x + threadIdx.x;
    if (t >= NUM_GRAPHS * NUM_CLASSES) return;
    int g = t / NUM_CLASSES, c = t - g * NUM_CLASSES;
    float inv = 1.0f / fmaxf(cnt[g], 1.0f);
    float acc = bfc[c];
    const float* pg = pooled + g * 256;
#pragma unroll 4
    for (int k = 0; k < 256; ++k)
        acc = fmaf(pg[k] * inv, Wfc[k * NUM_CLASSES + c], acc);
    out[t] = acc;
}

static inline int cdiv(long long a, int b) { return (int)((a + b - 1) / b); }

extern "C" void kernel_launch(void* const* d_in, const int* in_sizes, int n_in,
                              void* d_out, int out_size, void* d_ws, size_t ws_size,
                              hipStream_t stream) {
    const float* x    = (const float*)d_in[0];
    const int*   ei   = (const int*)  d_in[1];
    const int*   batch= (const int*)  d_in[2];
    const float* W1   = (const float*)d_in[3];
    const float* b1   = (const float*)d_in[4];
    const float* W2   = (const float*)d_in[5];
    const float* b2   = (const float*)d_in[6];
    const float* W3   = (const float*)d_in[7];
    const float* b3   = (const float*)d_in[8];
    const float* Wfc  = (const float*)d_in[9];
    const float* bfc  = (const float*)d_in[10];
    float* out = (float*)d_out;

    const int N = N_NODES, E = N_EDGES;
    const int* row = ei;         // edge_index[0]
    const int* col = ei + E;     // edge_index[1]

    char* ws = (char*)d_ws;
    float* bufG   = (float*)ws;                       // N*256 f32 (GEMM out)
    float* bufH   = (float*)(ws + 51200000);          // N*256 f32 (agg / relu)
    float* dinv   = (float*)(ws + 102400000);         // N f32
    float* pooled = (float*)(ws + 102600064);         // 16*256 f32
    float* cnt    = pooled + NUM_GRAPHS * 256;        // 16 f32

    const int B = 256;

    // --- degree + symmetric norm ---
    k_fill <<<cdiv(N, B), B, 0, stream>>>(dinv, 1.0f, N);          // self loop
    k_deg  <<<cdiv(E, B), B, 0, stream>>>(col, dinv, E);
    k_rsqrt<<<cdiv(N, B), B, 0, stream>>>(dinv, N);

    // --- layer 1: K=3 -> 64 ---
    k_gemm1    <<<cdiv((long long)N * 64, B), B, 0, stream>>>(x, W1, bufG, N);
    k_selfloop <<<cdiv((long long)N * 64, B), B, 0, stream>>>(bufG, dinv, bufH, N * 64, 6);
    k_edge     <<<cdiv((long long)E * 16, B), B, 0, stream>>>(bufG, row, col, dinv, bufH, E, 4, 64);
    k_bias_relu<<<cdiv((long long)N * 64, B), B, 0, stream>>>(bufH, b1, N * 64, 63);

    // --- layer 2: 64 -> 128 (WMMA) ---
    {
        int tiles = (N / 16) * (128 / 16);
        k_gemm_wmma<<<cdiv((long long)tiles * 32, B), B, 0, stream>>>(bufH, W2, bufG, N, 64, 128);
    }
    k_selfloop <<<cdiv((long long)N * 128, B), B, 0, stream>>>(bufG, dinv, bufH, N * 128, 7);
    k_edge     <<<cdiv((long long)E * 32, B), B, 0, stream>>>(bufG, row, col, dinv, bufH, E, 5, 128);
    k_bias_relu<<<cdiv((long long)N * 128, B), B, 0, stream>>>(bufH, b2, N * 128, 127);

    // --- layer 3: 128 -> 256 (WMMA) ---
    {
        int tiles = (N / 16) * (256 / 16);
        k_gemm_wmma<<<cdiv((long long)tiles * 32, B), B, 0, stream>>>(bufH, W3, bufG, N, 128, 256);
    }
    k_selfloop <<<cdiv((long long)N * 256, B), B, 0, stream>>>(bufG, dinv, bufH, N * 256, 8);
    k_edge     <<<cdiv((long long)E * 64, B), B, 0, stream>>>(bufG, row, col, dinv, bufH, E, 6, 256);
    k_bias_relu<<<cdiv((long long)N * 256, B), B, 0, stream>>>(bufH, b3, N * 256, 255);

    // --- mean pool + FC ---
    k_fill<<<cdiv(NUM_GRAPHS * 256 + NUM_GRAPHS, B), B, 0, stream>>>(pooled, 0.0f,
                                                                     NUM_GRAPHS * 256 + NUM_GRAPHS);
    k_pool<<<cdiv((long long)N * 256, B), B, 0, stream>>>(bufH, batch, pooled, cnt, N);
    k_fc  <<<cdiv(NUM_GRAPHS * NUM_CLASSES, B), B, 0, stream>>>(pooled, cnt, Wfc, bfc, out);
}